// OESM_CrossEntropy_30545807409163
// MI455X (gfx1250) — compile-verified
//
#include <hip/hip_runtime.h>
#include <math.h>

// ---------------------------------------------------------------------------
// CDNA5 async global->LDS staging (ASYNCcnt path), with safe fallback.
// Confirmed lowering: global_load_async_to_lds_b128 ... scale_offset
// ---------------------------------------------------------------------------
#ifndef __has_builtin
#define __has_builtin(x) 0
#endif

#if __has_builtin(__builtin_amdgcn_global_load_async_to_lds_b128)
#define USE_ASYNC_LDS 1
#else
#define USE_ASYNC_LDS 0
#endif

// Exact types for the builtin signature "vV4i*1V4i*3IiIi":
//   (AS1 int4*, AS3 int4*, imm offset, imm cpol)
typedef int v4i __attribute__((__vector_size__(4 * sizeof(int))));
typedef __attribute__((address_space(1))) v4i gas_v4i;  // global int4
typedef __attribute__((address_space(3))) v4i las_v4i;  // LDS int4

#if USE_ASYNC_LDS
#if __has_builtin(__builtin_amdgcn_s_wait_asynccnt)
#define WAIT_ASYNC(n) __builtin_amdgcn_s_wait_asynccnt(n)
#else
#define WAIT_ASYNC(n) asm volatile("s_wait_asynccnt %0" ::"i"(n) : "memory")
#endif
#else
#define WAIT_ASYNC(n) ((void)0)
#endif

#define NT 256     // threads per block (8 wave32)
#define NWAVE (NT / 32)
#define CH4 1024   // float4 elements per chunk buffer => 16 KB

// Stage one 16KB chunk of a row into an LDS buffer.
// Index clamping (instead of predication) keeps the async-instruction count
// uniform: exactly CH4/NT = 4 per wave per chunk, so ASYNCcnt bookkeeping is
// exact and s_wait_asynccnt 4 retires precisely the previous chunk.
__device__ __forceinline__ void stage_chunk(const float4* __restrict__ xr4,
                                            int C4, int chunk,
                                            float4* __restrict__ buf) {
  const int base = chunk * CH4;
#pragma unroll
  for (int k = 0; k < CH4 / NT; ++k) {
    const int i = (int)threadIdx.x + k * NT;
    int g = base + i;
    g = (g < C4) ? g : (C4 - 1);  // clamp: duplicate loads are harmless
#if USE_ASYNC_LDS
    __builtin_amdgcn_global_load_async_to_lds_b128(
        (gas_v4i*)(xr4 + g), (las_v4i*)(buf + i), 0, 0);
#else
    buf[i] = xr4[g];
#endif
  }
}

// Block-wide reduction tuned for wave32: 5 xor-shuffles intra-wave (no LDS,
// no barriers), one 8-slot LDS exchange, then every thread broadcast-reads
// the 8 partials so the result is uniform with just 2 barriers.
template <bool IS_MAX>
__device__ __forceinline__ float block_reduce(float v, float* red8, int tid) {
#pragma unroll
  for (int off = 16; off > 0; off >>= 1) {
    const float o = __shfl_xor(v, off, 32);
    v = IS_MAX ? fmaxf(v, o) : (v + o);
  }
  if ((tid & 31) == 0) red8[tid >> 5] = v;
  __syncthreads();
  float r = IS_MAX ? -__builtin_inff() : 0.0f;
#pragma unroll
  for (int w = 0; w < NWAVE; ++w) {
    const float p = red8[w];  // same address across lanes -> LDS broadcast
    r = IS_MAX ? fmaxf(r, p) : (r + p);
  }
  __syncthreads();  // red8 reusable afterwards
  return r;
}

// ---------------------------------------------------------------------------
// Kernel 1: per-row loss[i] = logsumexp(x[i,:]) - x[i, target[i]]
// One block per row; row streamed through a double-buffered LDS pipeline so
// every HBM byte is read exactly once and exp() overlaps the next fetch.
// ---------------------------------------------------------------------------
__global__ __launch_bounds__(NT) void OESM_row_loss_kernel(
    const float* __restrict__ x, const long long* __restrict__ target,
    float* __restrict__ loss, int C) {
  __shared__ float4 buf[2][CH4];  // 32 KB double buffer
  __shared__ float red8[NWAVE];

  const int row = blockIdx.x;
  const int tid = (int)threadIdx.x;
  const float* xr = x + (size_t)row * (size_t)C;
  const float4* xr4 = (const float4*)xr;
  const int C4 = C >> 2;  // C = 32000 -> 8000 float4
  const int nchunk = (C4 + CH4 - 1) / CH4;

  stage_chunk(xr4, C4, 0, &buf[0][0]);

  float m = -__builtin_inff();  // running max
  float s = 0.0f;               // running sum of exp(v - m)

  for (int c = 0; c < nchunk; ++c) {
    if (c + 1 < nchunk) {
      stage_chunk(xr4, C4, c + 1, &buf[(c + 1) & 1][0]);
      WAIT_ASYNC(4);  // retire chunk c; chunk c+1 (4 per wave) stays in flight
    } else {
      WAIT_ASYNC(0);
    }
    __syncthreads();  // chunk c fully resident in LDS for all waves

    const float4* b = &buf[c & 1][0];
    const int lim = min(C4 - c * CH4, CH4);  // = CH4 for all but last chunk

    // Pass 1: chunk max (from LDS)
    float lm = -__builtin_inff();
    if (lim == CH4) {
#pragma unroll
      for (int k = 0; k < CH4 / NT; ++k) {
        const float4 v = b[tid + k * NT];
        lm = fmaxf(lm, fmaxf(fmaxf(v.x, v.y), fmaxf(v.z, v.w)));
      }
    } else {
      for (int i = tid; i < lim; i += NT) {
        const float4 v = b[i];
        lm = fmaxf(lm, fmaxf(fmaxf(v.x, v.y), fmaxf(v.z, v.w)));
      }
    }
    const float cm = block_reduce<true>(lm, red8, tid);

    // Pass 2: chunk sum of exp(v - cm) (from LDS)
    float ls = 0.0f;
    if (lim == CH4) {
#pragma unroll
      for (int k = 0; k < CH4 / NT; ++k) {
        const float4 v = b[tid + k * NT];
        ls += __expf(v.x - cm) + __expf(v.y - cm) + __expf(v.z - cm) +
              __expf(v.w - cm);
      }
    } else {
      for (int i = tid; i < lim; i += NT) {
        const float4 v = b[i];
        ls += __expf(v.x - cm) + __expf(v.y - cm) + __expf(v.z - cm) +
              __expf(v.w - cm);
      }
    }
    const float cs = block_reduce<false>(ls, red8, tid);
    // block_reduce's trailing barrier fences all reads of buf[c&1] before the
    // next iteration restages into it.

    // Numerically-stable online combine (uniform across all threads)
    const float nm = fmaxf(m, cm);
    s = s * __expf(m - nm) + cs * __expf(cm - nm);
    m = nm;
  }

  if (tid == 0) {
    const long long t = target[row];
    const float xt = xr[(int)t];
    loss[row] = m + __logf(s) - xt;
  }
}

// ---------------------------------------------------------------------------
// Kernel 2: OHEM selection. Bitonic-sort B<=4096 losses ascending in LDS,
// then mean of ranks [lo, hi)  (lo = k1-k2, hi = k1).
// ---------------------------------------------------------------------------
#define NSEL 4096
__global__ __launch_bounds__(1024) void OESM_select_kernel(
    const float* __restrict__ loss, float* __restrict__ out, int B, int N,
    int lo, int hi) {
  __shared__ float vals[NSEL];
  __shared__ float red[32];
  const int tid = (int)threadIdx.x;

  for (int i = tid; i < N; i += 1024)
    vals[i] = (i < B) ? loss[i] : __builtin_inff();  // +inf pads sort to top
  __syncthreads();

  for (int k = 2; k <= N; k <<= 1) {
    for (int j = k >> 1; j > 0; j >>= 1) {
      for (int i = tid; i < N; i += 1024) {
        const int ixj = i ^ j;
        if (ixj > i) {
          const float a = vals[i];
          const float bb = vals[ixj];
          const bool up = ((i & k) == 0);
          if ((a > bb) == up) {
            vals[i] = bb;
            vals[ixj] = a;
          }
        }
      }
      __syncthreads();
    }
  }

  float sum = 0.0f;
  for (int i = lo + tid; i < hi; i += 1024) sum += vals[i];
#pragma unroll
  for (int off = 16; off > 0; off >>= 1) sum += __shfl_xor(sum, off, 32);
  if ((tid & 31) == 0) red[tid >> 5] = sum;
  __syncthreads();
  if (tid == 0) {
    float t = 0.0f;
#pragma unroll
    for (int w = 0; w < 32; ++w) t += red[w];
    out[0] = t / (float)(hi - lo);
  }
}

// ---------------------------------------------------------------------------
extern "C" void kernel_launch(void* const* d_in, const int* in_sizes, int n_in,
                              void* d_out, int out_size, void* d_ws,
                              size_t ws_size, hipStream_t stream) {
  (void)n_in;
  (void)out_size;
  (void)ws_size;
  const float* x = (const float*)d_in[0];
  const long long* target = (const long long*)d_in[1];  // int64 per reference
  const int B = in_sizes[1];             // 4096 rows
  const int C = in_sizes[0] / B;         // 32000 classes

  float* loss = (float*)d_ws;            // B floats of scratch

  OESM_row_loss_kernel<<<B, NT, 0, stream>>>(x, target, loss, C);

  // Match Python: k1 = int(0.9*B); k2 = int(0.7*k1)
  const int k1 = (int)(0.9 * (double)B);
  const int k2 = (int)(0.7 * (double)k1);
  const int lo = k1 - k2;
  const int hi = k1;
  int N = 1;
  while (N < B) N <<= 1;  // B=4096 -> N=4096 (fits NSEL LDS array)

  OESM_select_kernel<<<1, 1024, 0, stream>>>(loss, (float*)d_out, B, N, lo, hi);
}